// LocalAggregationLoss_1623497637975
// MI455X (gfx1250) — compile-verified
//
#include <hip/hip_runtime.h>
#include <math.h>

typedef __attribute__((ext_vector_type(2))) float v2f;
typedef __attribute__((ext_vector_type(8))) float v8f;

namespace {
constexpr int kDim   = 128;   // feature dim
constexpr int kNeigh = 200;   // neighbors per set
constexpr int kTiles = 13;    // ceil(200 / 16)
constexpr float kInvT = 1.0f / 0.07f;
}

__global__ __launch_bounds__(256)
void la_loss_kernel(const float* __restrict__ codes,
                    const float* __restrict__ bank,
                    const int*   __restrict__ idxBg,
                    const int*   __restrict__ idxCl,
                    float*       __restrict__ out)
{
    __shared__ __align__(16) float vLds[kDim];   // normalized code vector
    __shared__ int   idxLds[2][kNeigh + 8];      // staged neighbor indices (padded)
    __shared__ float partial[2][8];              // per-wave partial sums

    const int b    = blockIdx.x;
    const int tid  = threadIdx.x;
    const int wid  = tid >> 5;      // wave id (8 waves of 32)
    const int lane = tid & 31;
    const int m    = lane & 15;     // A-matrix row this lane owns
    const int half = lane >> 4;     // K-half (lanes 16-31 hold K=2,3 of each chunk)

    // ---- stage codes row + both index lists into LDS (coalesced) ----
    if (tid < kDim) vLds[tid] = codes[(size_t)b * kDim + tid];
    if (tid < kNeigh) {
        idxLds[0][tid] = idxBg[(size_t)b * kNeigh + tid];
        idxLds[1][tid] = idxCl[(size_t)b * kNeigh + tid];
    }
    __syncthreads();

    // ---- v = codes[b] / ||codes[b]||  (in place in LDS) ----
    if (wid == 0) {
        float ss = 0.f;
        #pragma unroll
        for (int i = 0; i < kDim; i += 32) {
            float c = vLds[i + lane];
            ss += c * c;
        }
        #pragma unroll
        for (int off = 16; off > 0; off >>= 1)
            ss += __shfl_xor(ss, off, 32);
        const float inv = 1.0f / sqrtf(ss);
        #pragma unroll
        for (int i = 0; i < kDim; i += 32)
            vLds[i + lane] *= inv;
    }
    __syncthreads();

    float sAcc[2] = {0.f, 0.f};

    // 26 tile-jobs (set, tile) striped across the 8 waves
    for (int job = wid; job < 2 * kTiles; job += 8) {
        const int set  = job & 1;
        const int tile = job >> 1;
        const int kRow = tile * 16 + m;
        const int kLd  = kRow < kNeigh ? kRow : kNeigh - 1;   // clamp for safe gather
        const float* __restrict__ row =
            bank + (size_t)idxLds[set][kLd] * kDim;           // ds_load_b32, no HBM stall

        // D = A(16 gathered rows x 128) * B(v broadcast to 16 cols), full FP32 WMMA.
        v8f accA = {}; v8f accB = {};
        #pragma unroll
        for (int c = 0; c < 32; c += 2) {       // 32 K-chunks of 4
            const int o0 = (c    ) * 4 + half * 2;
            const int o1 = (c + 1) * 4 + half * 2;
            v2f a0 = *reinterpret_cast<const v2f*>(row  + o0);
            v2f b0 = *reinterpret_cast<const v2f*>(vLds + o0);
            v2f a1 = *reinterpret_cast<const v2f*>(row  + o1);
            v2f b1 = *reinterpret_cast<const v2f*>(vLds + o1);
            accA = __builtin_amdgcn_wmma_f32_16x16x4_f32(
                       false, a0, false, b0, (short)0, accA, false, false);
            accB = __builtin_amdgcn_wmma_f32_16x16x4_f32(
                       false, a1, false, b1, (short)0, accB, false, false);
        }

        // Every column of D equals dot(row_m, v): VGPR r holds row r + 8*half.
        float s = 0.f;
        #pragma unroll
        for (int r = 0; r < 8; ++r) {
            const int k = tile * 16 + r + half * 8;
            const float d = (accA[r] + accB[r]) * kInvT;
            if (k < kNeigh) s += __expf(d);
        }
        sAcc[set] += s;
    }

    // lanes 0-15 saw rows 0-7 of each tile, lanes 16-31 rows 8-15 -> combine halves
    #pragma unroll
    for (int set = 0; set < 2; ++set) {
        const float tot = sAcc[set] + __shfl_xor(sAcc[set], 16, 32);
        if (lane == 0) partial[set][wid] = tot;
    }
    __syncthreads();

    if (tid == 0) {
        float d1 = 0.f, d2 = 0.f;
        #pragma unroll
        for (int w = 0; w < 8; ++w) { d1 += partial[0][w]; d2 += partial[1][w]; }
        out[b] = logf(d1) - logf(d2);
    }
}

extern "C" void kernel_launch(void* const* d_in, const int* in_sizes, int n_in,
                              void* d_out, int out_size, void* d_ws, size_t ws_size,
                              hipStream_t stream)
{
    const float* codes = (const float*)d_in[0];   // [B, 128] f32
    const float* bank  = (const float*)d_in[1];   // [1e6, 128] f32
    const int*   idxBg = (const int*)d_in[2];     // [B, 200] int
    const int*   idxCl = (const int*)d_in[3];     // [B, 200] int
    float* out = (float*)d_out;                   // [B] f32

    const int batch = in_sizes[0] / kDim;         // 1024
    la_loss_kernel<<<batch, 256, 0, stream>>>(codes, bank, idxBg, idxCl, out);
}